// ArcFaceHead_14319420965337
// MI455X (gfx1250) — compile-verified
//
#include <hip/hip_runtime.h>
#include <hip/hip_bf16.h>
#include <math.h>

// ArcFace head for MI455X (gfx1250, wave32).
// feat [B,512] fp32, W [2,512] fp32, label [B] int -> out [B,2] fp32.
// Memory-bound (~257MB @ 23.3TB/s ~= 11us). The feat@W^T part is mapped onto
// V_WMMA_F32_16X16X4_F32 (full fp32 precision); row norms ride along in VALU
// on the same A-fragment data (zero extra bandwidth).
//
// B-fragment columns >= 2 are sourced from a zero-filled LDS row instead of
// being masked with v_pk_mul: exact zeros, 2 fewer VALU ops per WMMA, and the
// WMMA can consume the ds_load results directly.

#if defined(__HIP_DEVICE_COMPILE__) && !__has_builtin(__builtin_amdgcn_wmma_f32_16x16x4_f32)
#error "missing __builtin_amdgcn_wmma_f32_16x16x4_f32 on device toolchain"
#endif

typedef __attribute__((ext_vector_type(2))) float v2f;
typedef __attribute__((ext_vector_type(4))) float v4f;
typedef __attribute__((ext_vector_type(8))) float v8f;

#define D_DIM 512
#define C_DIM 2
#define KC 64            // K chunk per staging round
#define TILE_STRIDE 68   // KC+4; stride ≡ 4 (mod 64) -> conflict-free frag reads
#define W_STRIDE 520     // ≡ 8 (mod 64) -> W rows land in different banks
#define W_ROWS 3         // rows 0,1 = classes; row 2 = zeros (for B cols >= 2)
#define WAVES 8
#define BLOCK (WAVES * 32)

__launch_bounds__(BLOCK)
__global__ void arcface_wmma_kernel(const float* __restrict__ feat,
                                    const float* __restrict__ W,
                                    const int* __restrict__ label,
                                    float* __restrict__ out,
                                    int numRows)
{
    __shared__ float wlds[W_ROWS * W_STRIDE];          // ~6.2 KB
    __shared__ float atile[WAVES][16 * TILE_STRIDE];   // ~34.8 KB
    __shared__ float outst[WAVES][16 * 2];             // 1 KB
    __shared__ float nacc[C_DIM];

    const int tid  = threadIdx.x;
    const int wave = tid >> 5;
    const int lane = tid & 31;

    if (tid < C_DIM) nacc[tid] = 0.0f;
    // Zero row 2 of wlds: source of exact-zero B columns.
    #pragma unroll
    for (int it = 0; it < D_DIM / BLOCK; ++it)
        wlds[2 * W_STRIDE + it * BLOCK + tid] = 0.0f;
    __syncthreads();

    // ---- Stage W into LDS and compute ||w0||^2, ||w1||^2 (once per block).
    {
        float s0 = 0.0f, s1 = 0.0f;
        #pragma unroll
        for (int it = 0; it < (C_DIM * D_DIM) / BLOCK; ++it) {
            int flat = it * BLOCK + tid;
            int row  = flat >> 9;          // D_DIM == 512
            int col  = flat & (D_DIM - 1);
            float w  = W[flat];
            wlds[row * W_STRIDE + col] = w;
            if (row == 0) s0 = fmaf(w, w, s0); else s1 = fmaf(w, w, s1);
        }
        atomicAdd(&nacc[0], s0);
        atomicAdd(&nacc[1], s1);
    }
    __syncthreads();

    const float wn0 = fmaxf(sqrtf(nacc[0]), 1e-12f);
    const float wn1 = fmaxf(sqrtf(nacc[1]), 1e-12f);

    // Lane roles in the WMMA fragments.
    const int col  = lane & 15;        // A: row index; B/D: column index
    const int hi   = lane >> 4;        // 0: K % 4 in {0,1}; 1: K % 4 in {2,3}
    const int koff = hi * 2;
    const int bsel = (col < C_DIM) ? col : 2;   // cols >= 2 read the zero row

    const int numTiles = numRows >> 4;
    const int gwave    = blockIdx.x * WAVES + wave;
    const int nwaves   = gridDim.x * WAVES;
    float* at = atile[wave];
    const float* wrow = &wlds[bsel * W_STRIDE + koff];

    for (int tile = gwave; tile < numTiles; tile += nwaves) {
        const int r0 = tile << 4;
        v8f  c  = {0.f, 0.f, 0.f, 0.f, 0.f, 0.f, 0.f, 0.f};
        float ss = 0.0f;                  // partial sum of squares (half-K of one row)

        for (int kc = 0; kc < D_DIM; kc += KC) {
            // Stage 16 x KC chunk: coalesced b128 global loads -> LDS tile.
            #pragma unroll
            for (int it = 0; it < (16 * KC / 4) / 32; ++it) {   // 8 iters
                int flat = it * 32 + lane;                       // float4 units
                int row  = flat >> 4;                            // KC/4 == 16 per row
                int c4   = flat & 15;
                v4f v = *(const v4f*)(feat + (size_t)(r0 + row) * D_DIM + kc + c4 * 4);
                *(v4f*)&at[row * TILE_STRIDE + c4 * 4] = v;
            }
            __builtin_amdgcn_wave_barrier();  // keep DS store->load program order

            // K loop: one V_WMMA_F32_16X16X4_F32 per 4 K values.
            #pragma unroll
            for (int k = 0; k < KC; k += 4) {
                v2f a = *(const v2f*)&at[col * TILE_STRIDE + k + koff];
                v2f b = *(const v2f*)&wrow[kc + k];
#if defined(__HIP_DEVICE_COMPILE__)
                c = __builtin_amdgcn_wmma_f32_16x16x4_f32(
                        false, a, false, b, (short)0, c, false, false);
#endif
                ss = fmaf(a.x, a.x, ss);
                ss = fmaf(a.y, a.y, ss);
            }
            __builtin_amdgcn_wave_barrier();
        }

        // Row norm: lanes l and l^16 hold complementary K-halves of row (l&15).
        ss += __shfl_xor(ss, 16, 32);

        // D layout: VGPR j, lanes 0-15 -> row j; lanes 16-31 -> row j+8.
        // Only columns 0,1 (lanes 0,1,16,17) carry real dot products.
        if (col < C_DIM) {
            #pragma unroll
            for (int j = 0; j < 8; ++j)
                outst[wave][(j + hi * 8) * 2 + col] = c[j];
        }
        // In-wave DS ordering makes the scatter visible to the next DS loads.
        if (lane < 16) {
            v2f d = *(const v2f*)&outst[wave][lane * 2];
            float xn = fmaxf(sqrtf(ss), 1e-12f);
            float c0 = d.x / (xn * wn0);
            float c1 = d.y / (xn * wn1);
            const float EPS = 1e-7f;
            c0 = fminf(fmaxf(c0, -1.0f + EPS), 1.0f - EPS);
            c1 = fminf(fmaxf(c1, -1.0f + EPS), 1.0f - EPS);
            // cos(arccos(c) + m) == c*cos(m) - sqrt(1-c^2)*sin(m)
            const float cosm = 0.8775825618903728f;   // cos(0.5)
            const float sinm = 0.479425538604203f;    // sin(0.5)
            float m0 = c0 * cosm - sqrtf(fmaxf(1.0f - c0 * c0, 0.0f)) * sinm;
            float m1 = c1 * cosm - sqrtf(fmaxf(1.0f - c1 * c1, 0.0f)) * sinm;
            int lbl = label[r0 + lane];
            v2f o;
            o.x = ((lbl == 0) ? m0 : c0) * 64.0f;
            o.y = ((lbl == 1) ? m1 : c1) * 64.0f;
            *(v2f*)(out + (size_t)(r0 + lane) * 2) = o;
        }
    }
}

extern "C" void kernel_launch(void* const* d_in, const int* in_sizes, int n_in,
                              void* d_out, int out_size, void* d_ws, size_t ws_size,
                              hipStream_t stream) {
    const float* feat  = (const float*)d_in[0];
    const float* W     = (const float*)d_in[1];
    const int*   label = (const int*)d_in[2];   // per harness ABI: integer -> const int*
    float*       out   = (float*)d_out;

    const int B = in_sizes[2];                  // 131072
    const int numTiles = B >> 4;                // 8192 tiles of 16 rows
    int blocks = (numTiles + WAVES - 1) / WAVES;
    if (blocks > 1024) blocks = 1024;           // grid-stride covers the rest
    if (blocks < 1) blocks = 1;

    arcface_wmma_kernel<<<blocks, BLOCK, 0, stream>>>(feat, W, label, out, B);
}